// GCN_7627861917726
// MI455X (gfx1250) — compile-verified
//
#include <hip/hip_runtime.h>
#include <math.h>

#define NN 100000
#define NE 1600000
#define D  128
#define NL 3

typedef __attribute__((ext_vector_type(2))) float v2f;
typedef __attribute__((ext_vector_type(8))) float v8f;

// ---------------- degree / normalization ----------------

__global__ void zero_f32_k(float* __restrict__ p, int n) {
    int i = blockIdx.x * blockDim.x + threadIdx.x;
    if (i < n) p[i] = 0.0f;
}

__global__ void deg_count_k(const int* __restrict__ col, float* __restrict__ deg, int e) {
    int i = blockIdx.x * blockDim.x + threadIdx.x;
    if (i < e) atomicAdd(&deg[col[i]], 1.0f);
}

__global__ void deg_inv_k(const float* __restrict__ deg, float* __restrict__ dinv, int n) {
    int i = blockIdx.x * blockDim.x + threadIdx.x;
    if (i < n) {
        float d = deg[i] + 1.0f;          // +1 self-loop; always > 0
        dinv[i] = rsqrtf(d);
    }
}

// ---------------- dense GEMM: H = SRC(NNxD) @ W(DxD), fp32 WMMA ----------------
// grid.x = NN/16 row panels, block = 256 (8 waves), wave w -> N-tile w.

__global__ __launch_bounds__(256) void gemm_wmma_k(const float* __restrict__ A,
                                                   const float* __restrict__ W,
                                                   float* __restrict__ H) {
    const int lane = threadIdx.x & 31;
    const int wave = threadIdx.x >> 5;
    const int m0   = blockIdx.x << 4;   // 16 rows per block
    const int n0   = wave << 4;         // 16 cols per wave
    const int half = lane >> 4;         // 0 or 1
    const int lm   = lane & 15;

    const float* __restrict__ arow = A + (size_t)(m0 + lm) * D;
    v8f c = {};

    #pragma unroll 4
    for (int k0 = 0; k0 < D; k0 += 4) {
        const int ka = k0 + 2 * half;
        // A 16x4: lane holds row m=lm, K = ka, ka+1 (8B contiguous load)
        v2f a = *(const v2f*)(arow + ka);
        // B 4x16: lane holds col n=lm, rows ka, ka+1
        v2f b;
        b.x = W[(size_t)ka * D + n0 + lm];
        b.y = W[(size_t)(ka + 1) * D + n0 + lm];
        c = __builtin_amdgcn_wmma_f32_16x16x4_f32(
            /*neg_a=*/false, a, /*neg_b=*/false, b,
            /*c_mod=*/(short)0, c, /*reuse_a=*/false, /*reuse_b=*/false);
    }

    // C/D 16x16: VGPR r -> M = r + 8*half, N = lm
    float* __restrict__ drow = H + (size_t)(m0 + 8 * half) * D + n0 + lm;
    #pragma unroll
    for (int r = 0; r < 8; ++r) drow[(size_t)r * D] = c[r];
}

// ---------------- self-loop init: acc = h * dinv^2 (full write, no atomics) ----------------
// one thread per float4; 32 float4 per node

__global__ void self_init_k(const float* __restrict__ h, const float* __restrict__ dinv,
                            float* __restrict__ acc) {
    int idx = blockIdx.x * blockDim.x + threadIdx.x;
    if (idx >= NN * (D / 4)) return;
    int node = idx >> 5;
    float s = dinv[node];
    s = s * s;
    float4 v = ((const float4*)h)[idx];
    float4 o;
    o.x = v.x * s; o.y = v.y * s; o.z = v.z * s; o.w = v.w * s;
    ((float4*)acc)[idx] = o;
}

// ---------------- edge scatter: acc[col] += h[row] * dinv[row]*dinv[col] ----------------
// one wave per edge (grid-stride); lane l handles dwords [4l, 4l+3]

__global__ __launch_bounds__(256) void edge_scatter_k(const int* __restrict__ row,
                                                      const int* __restrict__ col,
                                                      const float* __restrict__ dinv,
                                                      const float* __restrict__ h,
                                                      float* __restrict__ acc) {
    const int lane = threadIdx.x & 31;
    const int wid  = (blockIdx.x * blockDim.x + threadIdx.x) >> 5;
    const int nw   = (gridDim.x * blockDim.x) >> 5;
    for (int e = wid; e < NE; e += nw) {
        const int r = row[e];
        const int c = col[e];
        const float nrm = dinv[r] * dinv[c];
        const float4 v = ((const float4*)(h + (size_t)r * D))[lane];
        float* ap = acc + (size_t)c * D + (lane << 2);
        atomicAdd(ap + 0, v.x * nrm);
        atomicAdd(ap + 1, v.y * nrm);
        atomicAdd(ap + 2, v.z * nrm);
        atomicAdd(ap + 3, v.w * nrm);
    }
}

// ---------------- bias + relu (optionally in-place) ----------------

__global__ void bias_relu_k(const float* __restrict__ acc, const float* __restrict__ b,
                            float* __restrict__ out) {
    int idx = blockIdx.x * blockDim.x + threadIdx.x;
    if (idx >= NN * (D / 4)) return;
    float4 v  = ((const float4*)acc)[idx];
    float4 bb = ((const float4*)b)[idx & 31];
    float4 o;
    o.x = fmaxf(v.x + bb.x, 0.0f);
    o.y = fmaxf(v.y + bb.y, 0.0f);
    o.z = fmaxf(v.z + bb.z, 0.0f);
    o.w = fmaxf(v.w + bb.w, 0.0f);
    ((float4*)out)[idx] = o;
}

// ---------------- launch ----------------

extern "C" void kernel_launch(void* const* d_in, const int* in_sizes, int n_in,
                              void* d_out, int out_size, void* d_ws, size_t ws_size,
                              hipStream_t stream) {
    const float* x  = (const float*)d_in[0];
    const int*   ei = (const int*)d_in[1];
    const float* Ws = (const float*)d_in[2];
    const float* bs = (const float*)d_in[3];
    float* out = (float*)d_out;

    float* bufH   = (float*)d_ws;                  // NN*D
    float* bufAcc = bufH + (size_t)NN * D;         // NN*D
    float* deg    = bufAcc + (size_t)NN * D;       // NN
    float* dinv   = deg + NN;                      // NN

    const int* row = ei;
    const int* col = ei + NE;

    const int nvec = NN * (D / 4);                 // float4 elements per node buffer

    zero_f32_k<<<(NN + 255) / 256, 256, 0, stream>>>(deg, NN);
    deg_count_k<<<(NE + 255) / 256, 256, 0, stream>>>(col, deg, NE);
    deg_inv_k<<<(NN + 255) / 256, 256, 0, stream>>>(deg, dinv, NN);

    const float* src = x;
    for (int l = 0; l < NL; ++l) {
        gemm_wmma_k<<<NN / 16, 256, 0, stream>>>(src, Ws + (size_t)l * D * D, bufH);
        self_init_k<<<(nvec + 255) / 256, 256, 0, stream>>>(bufH, dinv, bufAcc);
        edge_scatter_k<<<8192, 256, 0, stream>>>(row, col, dinv, bufH, bufAcc);
        float* dst = (l == NL - 1) ? out : bufAcc;
        bias_relu_k<<<(nvec + 255) / 256, 256, 0, stream>>>(bufAcc, bs + (size_t)l * D, dst);
        src = bufAcc;
    }
}